// GConvGRU_76716705841717
// MI455X (gfx1250) — compile-verified
//
#include <hip/hip_runtime.h>
#include <hip/hip_bf16.h>
#include <math.h>

// ---------------------------------------------------------------------------
// GConvGRU (ChebConv K=2) for gfx1250.
//   SpMM (edge scatter) is the bandwidth-dominant part -> wave-per-edge,
//   scalarized edge state (SMEM) + float2 gathers + f32 atomic scatter.
//   Dense GEMMs use V_WMMA_F32_16X16X4_F32 (full fp32, matches reference).
// ---------------------------------------------------------------------------

typedef float v2f __attribute__((ext_vector_type(2)));
typedef float v8f __attribute__((ext_vector_type(8)));

#define IN_CH 8
#define OUT_CH 64

__device__ __forceinline__ v8f wmma_f32(v2f a, v2f b, v8f c) {
  // (neg_a, A, neg_b, B, c_mod, C, reuse_a, reuse_b)
  return __builtin_amdgcn_wmma_f32_16x16x4_f32(false, a, false, b, (short)0, c,
                                               false, false);
}

// Accumulate one K-block of a 16x16 output tile.
// A: row-major [*, ldA], rows rA (per-lane, lanes 0-15 == lanes 16-31 rows).
// W: row-major [Kdim, 64].
// Fragment layouts per CDNA5 ISA 7.12.2:
//   A 16x4 f32: v0 = K=k0 (lanes 0-15) / K=k0+2 (lanes 16-31), v1 = k0+1/k0+3
//   B 4x16 f32: v0 = row k0 cols 0..15 (lanes 0-15) / row k0+2 (lanes 16-31)
__device__ __forceinline__ v8f accum_block(const float* __restrict__ A, int ldA,
                                           int rA, const float* __restrict__ W,
                                           int Kdim, int colB, int half,
                                           v8f acc) {
#pragma unroll 4
  for (int k0 = 0; k0 < Kdim; k0 += 4) {
    int ka = k0 + 2 * half;
    v2f a;
    a.x = A[rA * ldA + ka];
    a.y = A[rA * ldA + ka + 1];
    v2f b;
    b.x = W[ka * OUT_CH + colB];
    b.y = W[(ka + 1) * OUT_CH + colB];
    acc = wmma_f32(a, b, acc);
  }
  return acc;
}

// ---------------------------------------------------------------------------

__global__ void __launch_bounds__(256) zero_kernel(float* p, int n) {
  int i = blockIdx.x * blockDim.x + threadIdx.x;
  if (i < n) p[i] = 0.0f;
}

__global__ void __launch_bounds__(256)
degree_kernel(const int* __restrict__ rowIdx, float* __restrict__ deg, int E) {
  int e = blockIdx.x * blockDim.x + threadIdx.x;
  if (e < E) atomicAdd(&deg[rowIdx[e]], 1.0f);
}

__global__ void __launch_bounds__(256) dinv_kernel(float* deg, int n) {
  int i = blockIdx.x * blockDim.x + threadIdx.x;
  if (i < n) {
    float d = deg[i];
    deg[i] = (d > 0.0f) ? rsqrtf(d) : 0.0f;
  }
}

// One wave (32 lanes) per edge. Edge id / indices / dinv values are
// wave-uniform -> force into SGPRs so they lower to SMEM loads. Each lane
// gathers one float2 (cols 2l, 2l+1) of the 64-wide block; lanes 0..7
// additionally cover the 8-wide block (when src8 != null).
__global__ void __launch_bounds__(256)
spmm_kernel(const int* __restrict__ rowIdx, const int* __restrict__ colIdx,
            const float* __restrict__ dinv, const float* __restrict__ src64,
            float* __restrict__ dst64, const float* __restrict__ src8,
            float* __restrict__ dst8, int E) {
  int e = __builtin_amdgcn_readfirstlane(
      (blockIdx.x * blockDim.x + threadIdx.x) >> 5);
  int lane = threadIdx.x & 31;
  if (e >= E) return;
  int r = __builtin_amdgcn_readfirstlane(rowIdx[e]);
  int c = __builtin_amdgcn_readfirstlane(colIdx[e]);
  float nrm = -dinv[r] * dinv[c];
  const float2* __restrict__ src = (const float2*)(src64 + (size_t)c * OUT_CH);
  float2 v = src[lane];  // global_load_b64, 8B aligned
  float* __restrict__ dst = dst64 + (size_t)r * OUT_CH + 2 * lane;
  atomicAdd(&dst[0], v.x * nrm);
  atomicAdd(&dst[1], v.y * nrm);
  if (src8 != nullptr && lane < IN_CH) {
    atomicAdd(&dst8[r * IN_CH + lane], src8[c * IN_CH + lane] * nrm);
  }
}

// Z / R gates: per block 16 rows x 64 cols, 4 waves (one 16-col tile each).
// Zpre = x@Wzx0 + Lx@Wzx1 + H@Wzh0 + LH@Wzh1 + bzx + bzh  (Rpre analogous)
__global__ void __launch_bounds__(128)
gates_wmma_kernel(const float* __restrict__ x, const float* __restrict__ H,
                  const float* __restrict__ Lx, const float* __restrict__ LH,
                  const float* __restrict__ Wzx, const float* __restrict__ Wzh,
                  const float* __restrict__ Wrx, const float* __restrict__ Wrh,
                  const float* __restrict__ bzx, const float* __restrict__ bzh,
                  const float* __restrict__ brx, const float* __restrict__ brh,
                  float* __restrict__ Zbuf, float* __restrict__ RH) {
  int lane = threadIdx.x & 31;
  int wv = threadIdx.x >> 5;
  int half = lane >> 4;
  int l15 = lane & 15;
  int row0 = blockIdx.x * 16;
  int col0 = wv * 16;
  int rA = row0 + l15;
  int colB = col0 + l15;

  v8f accZ = {};
  v8f accR = {};
  // K blocks: x (8), Lx (8), H (64), LH (64); W[k] at offset k*d*64.
  accZ = accum_block(x, IN_CH, rA, Wzx, IN_CH, colB, half, accZ);
  accR = accum_block(x, IN_CH, rA, Wrx, IN_CH, colB, half, accR);
  accZ = accum_block(Lx, IN_CH, rA, Wzx + IN_CH * OUT_CH, IN_CH, colB, half, accZ);
  accR = accum_block(Lx, IN_CH, rA, Wrx + IN_CH * OUT_CH, IN_CH, colB, half, accR);
  accZ = accum_block(H, OUT_CH, rA, Wzh, OUT_CH, colB, half, accZ);
  accR = accum_block(H, OUT_CH, rA, Wrh, OUT_CH, colB, half, accR);
  accZ = accum_block(LH, OUT_CH, rA, Wzh + OUT_CH * OUT_CH, OUT_CH, colB, half, accZ);
  accR = accum_block(LH, OUT_CH, rA, Wrh + OUT_CH * OUT_CH, OUT_CH, colB, half, accR);

  float bZ = bzx[colB] + bzh[colB];
  float bR = brx[colB] + brh[colB];
#pragma unroll
  for (int i = 0; i < 8; ++i) {
    int r = row0 + i + 8 * half;  // C/D layout: VGPR i -> row i (+8 upper half)
    float z = 1.0f / (1.0f + __expf(-(accZ[i] + bZ)));
    float rg = 1.0f / (1.0f + __expf(-(accR[i] + bR)));
    Zbuf[r * OUT_CH + colB] = z;
    RH[r * OUT_CH + colB] = rg * H[r * OUT_CH + colB];
  }
}

// Candidate + GRU blend:
// Cpre = x@Wcx0 + Lx@Wcx1 + RH@Wch0 + LRH@Wch1 + bcx + bch
// out  = Z*H + (1-Z)*tanh(Cpre)
__global__ void __launch_bounds__(128)
cand_out_wmma_kernel(const float* __restrict__ x, const float* __restrict__ H,
                     const float* __restrict__ Lx, const float* __restrict__ RH,
                     const float* __restrict__ LRH,
                     const float* __restrict__ Zbuf,
                     const float* __restrict__ Wcx,
                     const float* __restrict__ Wch,
                     const float* __restrict__ bcx,
                     const float* __restrict__ bch, float* __restrict__ out) {
  int lane = threadIdx.x & 31;
  int wv = threadIdx.x >> 5;
  int half = lane >> 4;
  int l15 = lane & 15;
  int row0 = blockIdx.x * 16;
  int col0 = wv * 16;
  int rA = row0 + l15;
  int colB = col0 + l15;

  v8f accC = {};
  accC = accum_block(x, IN_CH, rA, Wcx, IN_CH, colB, half, accC);
  accC = accum_block(Lx, IN_CH, rA, Wcx + IN_CH * OUT_CH, IN_CH, colB, half, accC);
  accC = accum_block(RH, OUT_CH, rA, Wch, OUT_CH, colB, half, accC);
  accC = accum_block(LRH, OUT_CH, rA, Wch + OUT_CH * OUT_CH, OUT_CH, colB, half, accC);

  float bC = bcx[colB] + bch[colB];
#pragma unroll
  for (int i = 0; i < 8; ++i) {
    int r = row0 + i + 8 * half;
    float cv = tanhf(accC[i] + bC);
    float z = Zbuf[r * OUT_CH + colB];
    float h = H[r * OUT_CH + colB];
    out[r * OUT_CH + colB] = z * h + (1.0f - z) * cv;
  }
}

// ---------------------------------------------------------------------------

extern "C" void kernel_launch(void* const* d_in, const int* in_sizes, int n_in,
                              void* d_out, int out_size, void* d_ws,
                              size_t ws_size, hipStream_t stream) {
  const float* x = (const float*)d_in[0];
  const float* H = (const float*)d_in[1];
  const int* edge = (const int*)d_in[2];
  const float* Wzx = (const float*)d_in[3];
  const float* bzx = (const float*)d_in[4];
  const float* Wzh = (const float*)d_in[5];
  const float* bzh = (const float*)d_in[6];
  const float* Wrx = (const float*)d_in[7];
  const float* brx = (const float*)d_in[8];
  const float* Wrh = (const float*)d_in[9];
  const float* brh = (const float*)d_in[10];
  const float* Wcx = (const float*)d_in[11];
  const float* bcx = (const float*)d_in[12];
  const float* Wch = (const float*)d_in[13];
  const float* bch = (const float*)d_in[14];

  const int N = in_sizes[0] / IN_CH;
  const int E = in_sizes[2] / 2;
  const int* rowIdx = edge;
  const int* colIdx = edge + E;

  // Workspace layout (floats):
  float* ws = (float*)d_ws;
  float* dinv = ws;                    // [0, N)        deg -> dinv in place
  float* Lx = dinv + N;                // N*8
  float* LH = Lx + (size_t)N * IN_CH;  // N*64
  float* LRH = LH + (size_t)N * OUT_CH;
  float* Zbuf = LRH + (size_t)N * OUT_CH;
  float* RH = Zbuf + (size_t)N * OUT_CH;
  const int nZero = N * (1 + IN_CH + OUT_CH + OUT_CH);  // dinv, Lx, LH, LRH

  // 1) zero accumulators
  zero_kernel<<<(nZero + 255) / 256, 256, 0, stream>>>(dinv, nZero);
  // 2) degree
  degree_kernel<<<(E + 255) / 256, 256, 0, stream>>>(rowIdx, dinv, E);
  // 3) dinv = rsqrt(deg)
  dinv_kernel<<<(N + 255) / 256, 256, 0, stream>>>(dinv, N);
  // 4) Lx = Â x, LH = Â H   (wave per edge)
  {
    long long threads = (long long)E * 32;
    spmm_kernel<<<(int)((threads + 255) / 256), 256, 0, stream>>>(
        rowIdx, colIdx, dinv, H, LH, x, Lx, E);
  }
  // 5) gates: Z, R⊙H
  gates_wmma_kernel<<<N / 16, 128, 0, stream>>>(x, H, Lx, LH, Wzx, Wzh, Wrx,
                                                Wrh, bzx, bzh, brx, brh, Zbuf,
                                                RH);
  // 6) LRH = Â (R⊙H)
  {
    long long threads = (long long)E * 32;
    spmm_kernel<<<(int)((threads + 255) / 256), 256, 0, stream>>>(
        rowIdx, colIdx, dinv, RH, LRH, nullptr, nullptr, E);
  }
  // 7) candidate + blend -> out
  cand_out_wmma_kernel<<<N / 16, 128, 0, stream>>>(
      x, H, Lx, RH, LRH, Zbuf, Wcx, Wch, bcx, bch, (float*)d_out);
}